// AdaptiveEmbedding_60138132078897
// MI455X (gfx1250) — compile-verified
//
#include <hip/hip_runtime.h>
#include <hip/hip_bf16.h>

#define CUT0 20000
#define CUT1 60000
#define DOUT 1024
#define NTOK 32768

typedef __attribute__((ext_vector_type(2))) float v2f;
typedef __attribute__((ext_vector_type(8))) float v8f;

// ---------------------------------------------------------------------------
// Kernel 1: reset compaction counters (must happen every launch: graph replays
// do not re-zero the workspace).
__global__ void init_kernel(int* __restrict__ ctr) {
    if (threadIdx.x < 2) ctr[threadIdx.x] = 0;
}

// ---------------------------------------------------------------------------
// Kernel 2: classify tokens into cluster-1 / cluster-2 compacted lists.
// Block-local LDS compaction + one global atomic per (block, cluster).
// List order is nondeterministic but the final output is scatter-by-token,
// so d_out is deterministic.
__global__ void classify_kernel(const int* __restrict__ ids,
                                int* __restrict__ ctr,
                                int* __restrict__ tok1,
                                int* __restrict__ tok2) {
    __shared__ int lc[2];
    __shared__ int gb[2];
    const int t = blockIdx.x * 256 + threadIdx.x;
    if (threadIdx.x < 2) lc[threadIdx.x] = 0;
    __syncthreads();
    const int id = ids[t];
    int p1 = -1, p2 = -1;
    if (id >= CUT0 && id < CUT1) p1 = atomicAdd(&lc[0], 1);
    else if (id >= CUT1)         p2 = atomicAdd(&lc[1], 1);
    __syncthreads();
    if (threadIdx.x < 2) gb[threadIdx.x] = atomicAdd(&ctr[threadIdx.x], lc[threadIdx.x]);
    __syncthreads();
    if (p1 >= 0) tok1[gb[0] + p1] = t;
    if (p2 >= 0) tok2[gb[1] + p2] = t;
}

// ---------------------------------------------------------------------------
// Kernel 3: cluster 0 — direct row copy emb0[id] -> out[t].  One wave per
// token, float4 (global_load_b128 / global_store_b128) fully coalesced.
__global__ void copy0_kernel(const int* __restrict__ ids,
                             const float* __restrict__ emb0,
                             float* __restrict__ out) {
    const int t    = blockIdx.x * 8 + (threadIdx.x >> 5);
    const int lane = threadIdx.x & 31;
    const int id   = ids[t];
    if (id >= CUT0) return;
    const float4* __restrict__ src = (const float4*)(emb0 + (size_t)id * DOUT);
    float4* __restrict__       dst = (float4*)(out + (size_t)t * DOUT);
#pragma unroll
    for (int i = 0; i < 8; ++i)
        dst[lane + i * 32] = src[lane + i * 32];
}

// ---------------------------------------------------------------------------
// Kernel 4: compacted projection GEMM  C[n, 0:1024] = A[n, 0:K] @ W^T + bias,
// rows scattered back to out[token].  Block = 256 threads = 8 waves.
// Block tile: 16 tokens x 128 columns; each wave: one 16x16 f32 WMMA tile.
// grid = (8 column-blocks, 2048 M-tiles); blocks past the compacted count
// exit uniformly (EXEC stays all-ones at every WMMA).
template <int K>
__global__ void gemm_proj_kernel(const int* __restrict__ ids,
                                 const float* __restrict__ emb,
                                 const float* __restrict__ w,
                                 const float* __restrict__ bias,
                                 const int* __restrict__ tok,
                                 const int* __restrict__ cnt,
                                 float* __restrict__ out,
                                 int cut) {
    constexpr int LDA = K + 4;              // +4 dwords: bank spread, keeps 16B align
    __shared__ float Abuf[16 * LDA];
    __shared__ int   stok[16];
    __shared__ int   sloc[16];

    const int n  = cnt[0];
    const int mt = blockIdx.y;
    if (mt * 16 >= n) return;               // block-uniform early exit

    // --- stage token ids / local embedding rows for this M-tile ------------
    if (threadIdx.x < 16) {
        const int gr = mt * 16 + threadIdx.x;
        const int tr = (gr < n) ? tok[gr] : -1;
        stok[threadIdx.x] = tr;
        sloc[threadIdx.x] = (tr >= 0) ? (ids[tr] - cut) : -1;
    }
    __syncthreads();

    // --- gather A tile (16 x K) into LDS, coalesced float4 -----------------
    {
        const int r   = threadIdx.x >> 4;    // row 0..15
        const int c4  = threadIdx.x & 15;    // float4 group within sweep
        const int loc = sloc[r];
        const float* __restrict__ srow = emb + (size_t)(loc < 0 ? 0 : loc) * K;
#pragma unroll
        for (int s = 0; s < K / 64; ++s) {
            const int c = (c4 + s * 16) * 4;
            float4 v = make_float4(0.f, 0.f, 0.f, 0.f);
            if (loc >= 0) v = *(const float4*)(srow + c);
            *(float4*)&Abuf[r * LDA + c] = v;
        }
    }
    __syncthreads();

    // --- WMMA K-loop: D = A(16xK) x W^T(Kx16) ------------------------------
    const int lane = threadIdx.x & 31;
    const int wv   = threadIdx.x >> 5;
    const int n0   = blockIdx.x * 128 + wv * 16;   // this wave's 16 columns
    const int nl   = lane & 15;                    // N col (B/C/D) == M row (A)
    const int kh   = (lane >> 4) * 2;              // K sub-offset within 4-step

    const float* __restrict__ wrow = w + (size_t)(n0 + nl) * K + kh;  // 8B aligned
    const float* __restrict__ arow = Abuf + nl * LDA + kh;            // 8B aligned

    v8f acc0 = {};
    v8f acc1 = {};
#pragma unroll
    for (int k = 0; k < K; k += 8) {
        v2f a0 = *(const v2f*)(arow + k);
        v2f b0 = *(const v2f*)(wrow + k);
        acc0 = __builtin_amdgcn_wmma_f32_16x16x4_f32(false, a0, false, b0,
                                                     (short)0, acc0, false, false);
        v2f a1 = *(const v2f*)(arow + k + 4);
        v2f b1 = *(const v2f*)(wrow + k + 4);
        acc1 = __builtin_amdgcn_wmma_f32_16x16x4_f32(false, a1, false, b1,
                                                     (short)0, acc1, false, false);
    }

    // --- scatter rows back to out[token] -----------------------------------
    const float bv   = bias[n0 + nl];
    const int   mrow = (lane >> 4) * 8;    // C/D: vgpr j -> row j (+8 for hi lanes)
    const int   col  = n0 + nl;
#pragma unroll
    for (int j = 0; j < 8; ++j) {
        const int tr = stok[mrow + j];
        if (tr >= 0)
            out[(size_t)tr * DOUT + col] = acc0[j] + acc1[j] + bv;
    }
}

// ---------------------------------------------------------------------------
extern "C" void kernel_launch(void* const* d_in, const int* in_sizes, int n_in,
                              void* d_out, int out_size, void* d_ws, size_t ws_size,
                              hipStream_t stream) {
    const int*   ids  = (const int*)d_in[0];
    const float* emb0 = (const float*)d_in[1];
    const float* emb1 = (const float*)d_in[2];
    const float* emb2 = (const float*)d_in[3];
    const float* w1   = (const float*)d_in[4];
    const float* b1   = (const float*)d_in[5];
    const float* w2   = (const float*)d_in[6];
    const float* b2   = (const float*)d_in[7];
    float* out = (float*)d_out;

    // workspace: [0,8)   two int counters (n1, n2)
    //            [256, 256+128K)        tok1 list
    //            [256+128K, 256+256K)   tok2 list
    int* ctr  = (int*)d_ws;
    int* tok1 = (int*)((char*)d_ws + 256);
    int* tok2 = (int*)((char*)d_ws + 256 + NTOK * sizeof(int));

    init_kernel<<<1, 32, 0, stream>>>(ctr);
    classify_kernel<<<NTOK / 256, 256, 0, stream>>>(ids, ctr, tok1, tok2);
    copy0_kernel<<<NTOK / 8, 256, 0, stream>>>(ids, emb0, out);
    gemm_proj_kernel<256><<<dim3(8, NTOK / 16), 256, 0, stream>>>(
        ids, emb1, w1, b1, tok1, ctr + 0, out, CUT0);
    gemm_proj_kernel<64><<<dim3(8, NTOK / 16), 256, 0, stream>>>(
        ids, emb2, w2, b2, tok2, ctr + 1, out, CUT1);
}